// Recurrent_GCN_5385888989804
// MI455X (gfx1250) — compile-verified
//
#include <hip/hip_runtime.h>
#include <hip/hip_bf16.h>
#include <math.h>

#define NUM_NODES 20000
#define LOOKBACK  7
#define DAYS      4
#define NFEAT     5
#define XCH       (LOOKBACK * NFEAT)   /* 35 feature channels per node-day */
#define HIDDEN    128
#define NUM_EDGES 320000
#define MTOT      (DAYS * NUM_NODES)   /* 80000 rows, all days batched */

typedef __attribute__((ext_vector_type(2))) float v2f;
typedef __attribute__((ext_vector_type(8))) float v8f;

__device__ __forceinline__ float sigmoidf_(float x) { return 1.0f / (1.0f + expf(-x)); }

// ================================================================ one-time graph setup
__global__ void k_deg_int(const int* __restrict__ dst, int* __restrict__ cnt) {
    int e = blockIdx.x * 256 + threadIdx.x;
    if (e < NUM_EDGES) atomicAdd(&cnt[dst[e]], 1);
}

__global__ void k_node_norm(const int* __restrict__ cnt, float* __restrict__ dinv,
                            float* __restrict__ selfn) {
    int n = blockIdx.x * 256 + threadIdx.x;
    if (n < NUM_NODES) {
        float d = (float)cnt[n] + 1.0f;          // self-loop included
        dinv[n]  = rsqrtf(d);
        selfn[n] = 1.0f / d;
    }
}

__global__ void k_edge_norm(const int* __restrict__ src, const int* __restrict__ dst,
                            const float* __restrict__ dinv, float* __restrict__ nrme) {
    int e = blockIdx.x * 256 + threadIdx.x;
    if (e < NUM_EDGES) nrme[e] = dinv[src[e]] * dinv[dst[e]];
}

// single-block exclusive scan of cnt[N] -> rowptr[N+1]  (256 threads, 79 elems each)
#define SCAN_CHUNK 79     /* 256*79 = 20224 >= NUM_NODES */
__global__ void k_scan(const int* __restrict__ cnt, int* __restrict__ rowptr) {
    __shared__ int part[256];
    int tid = threadIdx.x;
    int begin = tid * SCAN_CHUNK;
    int end = begin + SCAN_CHUNK; if (end > NUM_NODES) end = NUM_NODES;
    int s = 0;
    for (int i = begin; i < end && i < NUM_NODES; ++i) s += cnt[i];
    part[tid] = s;
    __syncthreads();
    for (int off = 1; off < 256; off <<= 1) {
        int v = 0;
        if (tid >= off) v = part[tid - off];
        __syncthreads();
        if (tid >= off) part[tid] += v;
        __syncthreads();
    }
    int run = (tid == 0) ? 0 : part[tid - 1];
    for (int i = begin; i < end && i < NUM_NODES; ++i) { rowptr[i] = run; run += cnt[i]; }
    if (tid == 255) rowptr[NUM_NODES] = run;     // == NUM_EDGES
}

__global__ void k_fill_csr(const int* __restrict__ src, const int* __restrict__ dst,
                           const float* __restrict__ nrme, const int* __restrict__ rowptr,
                           int* __restrict__ cursor, int* __restrict__ csr_eid,
                           int* __restrict__ csr_src, float* __restrict__ csr_w) {
    int e = blockIdx.x * 256 + threadIdx.x;
    if (e >= NUM_EDGES) return;
    int d = dst[e];
    int slot = rowptr[d] + atomicAdd(&cursor[d], 1);
    csr_eid[slot] = e;
    csr_src[slot] = src[e];
    csr_w[slot]   = nrme[e];
}

// per-node insertion sort by edge id -> deterministic accumulation order every call
__global__ void k_sort_csr(const int* __restrict__ rowptr, int* __restrict__ eid,
                           int* __restrict__ srcs, float* __restrict__ ws) {
    int n = blockIdx.x * 256 + threadIdx.x;
    if (n >= NUM_NODES) return;
    int beg = rowptr[n], end = rowptr[n + 1];
    for (int i = beg + 1; i < end; ++i) {
        int ke = eid[i]; int ks = srcs[i]; float kw = ws[i];
        int j = i - 1;
        while (j >= beg && eid[j] > ke) {
            eid[j + 1] = eid[j]; srcs[j + 1] = srcs[j]; ws[j + 1] = ws[j]; --j;
        }
        eid[j + 1] = ke; srcs[j + 1] = ks; ws[j + 1] = kw;
    }
}

// xa = A_hat * x over all 35 raw channels, once per launch (layer-1 agg commuted past W1)
__global__ void k_agg_x(const float* __restrict__ x, const int* __restrict__ rowptr,
                        const int* __restrict__ csr_src, const float* __restrict__ csr_w,
                        const float* __restrict__ selfn, float* __restrict__ xa) {
    int wid  = blockIdx.x * (blockDim.x >> 5) + (threadIdx.x >> 5);   // (node, day)
    int lane = threadIdx.x & 31;
    if (wid >= MTOT) return;
    int day = wid / NUM_NODES, n = wid % NUM_NODES;
    size_t rbase = (size_t)day * NUM_NODES * XCH;
    const float* xr = x + rbase + (size_t)n * XCH;
    float sn = selfn[n];
    int beg = rowptr[n], end = rowptr[n + 1];
    for (int c = lane; c < XCH; c += 32) {
        float acc = xr[c] * sn;
        for (int i = beg; i < end; ++i)
            acc += x[rbase + (size_t)csr_src[i] * XCH + c] * csr_w[i];
        xa[(size_t)wid * XCH + c] = acc;
    }
}

// layer-1 inside time loop: out = xa_t @ W1 + b1   (K=5, scalar; relu fused into next GEMM)
__global__ void k_gcn1(const float* __restrict__ xa, const float* __restrict__ W1,
                       const float* __restrict__ b1, float* __restrict__ out, int t) {
    int idx = blockIdx.x * 256 + threadIdx.x;       // over MTOT*128
    int m = idx >> 7, c = idx & (HIDDEN - 1);
    const float* xr = xa + (size_t)m * XCH + t * NFEAT;
    float s = b1[c];
#pragma unroll
    for (int f = 0; f < NFEAT; ++f) s = fmaf(xr[f], W1[f * HIDDEN + c], s);
    out[idx] = s;
}

// 128-channel CSR aggregation: agg = selfn*h + bias + sum_e w*h[src]  (no atomics)
__global__ void k_aggregate(const float* __restrict__ h, const int* __restrict__ rowptr,
                            const int* __restrict__ csr_src, const float* __restrict__ csr_w,
                            const float* __restrict__ selfn, const float* __restrict__ bias,
                            float* __restrict__ agg) {
    int wid  = blockIdx.x * (blockDim.x >> 5) + (threadIdx.x >> 5);   // (node, day)
    int lane = threadIdx.x & 31;
    if (wid >= MTOT) return;
    int day = wid / NUM_NODES, n = wid % NUM_NODES;
    size_t base = (size_t)day * NUM_NODES * HIDDEN;
    int c = lane << 2;                                // 4 channels per lane (b128 loads)
    float sn = selfn[n];
    float4 acc = *(const float4*)&h[base + ((size_t)n << 7) + c];
    float4 bb  = *(const float4*)&bias[c];
    acc.x = acc.x * sn + bb.x;  acc.y = acc.y * sn + bb.y;
    acc.z = acc.z * sn + bb.z;  acc.w = acc.w * sn + bb.w;
    int beg = rowptr[n], end = rowptr[n + 1];
    for (int i = beg; i < end; ++i) {                 // uniform scalar loads
        int s   = csr_src[i];
        float w = csr_w[i];
        float4 hv = *(const float4*)&h[base + ((size_t)s << 7) + c];
        acc.x = fmaf(hv.x, w, acc.x);  acc.y = fmaf(hv.y, w, acc.y);
        acc.z = fmaf(hv.z, w, acc.z);  acc.w = fmaf(hv.w, w, acc.w);
    }
    *(float4*)&agg[base + ((size_t)n << 7) + c] = acc;
}

// ---------------------------------------------------------------- fp32 WMMA GEMM
// C[M,Ncol] = op(A[M,K]) * B + bias ; B addressed as B[k*ldbk + j*ldbn]
template <bool RELU_A>
__global__ void k_wmma_gemm(const float* __restrict__ A, const float* __restrict__ B,
                            const float* __restrict__ bias, float* __restrict__ C,
                            int M, int Ncol, int K, int lda, int ldbk, int ldbn, int ldc) {
    const int wave = blockIdx.x * (blockDim.x >> 5) + (threadIdx.x >> 5);
    const int lane = threadIdx.x & 31;
    const int ntn  = Ncol >> 4;
    const int m0 = (wave / ntn) << 4;
    const int n0 = (wave % ntn) << 4;
    if (m0 >= M) return;                              // wave-uniform; EXEC stays all-1

    const int hl = lane >> 4;                         // K half select (ISA A/B layout)
    const int l  = lane & 15;                         // row (A) / col (B)
    const float* Ar = A + (size_t)(m0 + l) * lda;
    const float* Bc = B + (size_t)(n0 + l) * ldbn;

    v8f acc = {};
    for (int k = 0; k < K; k += 4) {
        const int ka = k + (hl << 1);                 // lanes 16-31 carry K+2,K+3
        float a0 = Ar[ka], a1 = Ar[ka + 1];
        if (RELU_A) { a0 = fmaxf(a0, 0.0f); a1 = fmaxf(a1, 0.0f); }
        v2f av; av.x = a0; av.y = a1;
        v2f bv; bv.x = Bc[(size_t)ka * ldbk]; bv.y = Bc[(size_t)(ka + 1) * ldbk];
        acc = __builtin_amdgcn_wmma_f32_16x16x4_f32(false, av, false, bv,
                                                    (short)0, acc, false, false);
    }

    const int col = n0 + l;
    const float bb = bias ? bias[col] : 0.0f;
    float* Cp = C + (size_t)(m0 + (hl << 3)) * ldc + col;
#pragma unroll
    for (int i = 0; i < 8; ++i) Cp[(size_t)i * ldc] = acc[i] + bb;
}

// ---------------------------------------------------------------- GRU gates
__global__ void k_gru_gates(const float* __restrict__ gi, const float* __restrict__ gh,
                            float* __restrict__ h) {
    int idx = blockIdx.x * 256 + threadIdx.x;       // over MTOT*128
    int m = idx >> 7, c = idx & (HIDDEN - 1);
    const float* gim = gi + (size_t)m * (3 * HIDDEN);
    const float* ghm = gh + (size_t)m * (3 * HIDDEN);
    float r = sigmoidf_(gim[c] + ghm[c]);
    float z = sigmoidf_(gim[HIDDEN + c] + ghm[HIDDEN + c]);
    float n = tanhf(gim[2 * HIDDEN + c] + r * ghm[2 * HIDDEN + c]);
    h[idx] = (1.0f - z) * n + z * h[idx];
}

// ---------------------------------------------------------------- output head + softmax
__global__ void k_head(const float* __restrict__ h, const float* __restrict__ Wo,
                       const float* __restrict__ bo, float* __restrict__ out) {
    int m = blockIdx.x * 256 + threadIdx.x;
    if (m >= MTOT) return;
    const float* row = h + (size_t)m * HIDDEN;
    float acc[5];
#pragma unroll
    for (int j = 0; j < 5; ++j) acc[j] = bo[j];
    for (int k = 0; k < HIDDEN; ++k) {
        float hv = row[k];
#pragma unroll
        for (int j = 0; j < 5; ++j) acc[j] = fmaf(hv, Wo[k * 5 + j], acc[j]);
    }
    float mx = acc[0];
#pragma unroll
    for (int j = 1; j < 5; ++j) mx = fmaxf(mx, acc[j]);
    float s = 0.0f;
#pragma unroll
    for (int j = 0; j < 5; ++j) { acc[j] = expf(acc[j] - mx); s += acc[j]; }
    float inv = 1.0f / s;
#pragma unroll
    for (int j = 0; j < 5; ++j) out[(size_t)m * 5 + j] = acc[j] * inv;
}

// ================================================================ launch
extern "C" void kernel_launch(void* const* d_in, const int* in_sizes, int n_in,
                              void* d_out, int out_size, void* d_ws, size_t ws_size,
                              hipStream_t stream) {
    (void)in_sizes; (void)n_in; (void)out_size; (void)ws_size;
    const float* x     = (const float*)d_in[0];
    const int*   ei    = (const int*)d_in[1];
    const float* W1    = (const float*)d_in[2];
    const float* b1    = (const float*)d_in[3];
    const float* W2    = (const float*)d_in[4];
    const float* b2    = (const float*)d_in[5];
    const float* W_ih  = (const float*)d_in[6];
    const float* W_hh  = (const float*)d_in[7];
    const float* b_ih  = (const float*)d_in[8];
    const float* b_hh  = (const float*)d_in[9];
    const float* W_out = (const float*)d_in[10];
    const float* b_out = (const float*)d_in[11];
    float* out = (float*)d_out;

    const int* src = ei;
    const int* dst = ei + NUM_EDGES;

    // 256B-aligned workspace carve-out
    char* p = (char*)d_ws;
    auto alloc = [&](size_t bytes) -> void* {
        void* r = (void*)p;
        p += (bytes + 255) & ~(size_t)255;
        return r;
    };
    int*   cnt     = (int*)  alloc(NUM_NODES * sizeof(int));
    int*   rowptr  = (int*)  alloc((NUM_NODES + 1) * sizeof(int));
    int*   cursor  = (int*)  alloc(NUM_NODES * sizeof(int));
    int*   csr_eid = (int*)  alloc(NUM_EDGES * sizeof(int));
    int*   csr_src = (int*)  alloc(NUM_EDGES * sizeof(int));
    float* csr_w   = (float*)alloc(NUM_EDGES * sizeof(float));
    float* dinv    = (float*)alloc(NUM_NODES * sizeof(float));
    float* selfn   = (float*)alloc(NUM_NODES * sizeof(float));
    float* nrme    = (float*)alloc(NUM_EDGES * sizeof(float));
    float* xa      = (float*)alloc((size_t)MTOT * XCH * sizeof(float));
    float* bufH    = (float*)alloc((size_t)MTOT * HIDDEN * sizeof(float));
    float* bufA    = (float*)alloc((size_t)MTOT * HIDDEN * sizeof(float));
    float* state   = (float*)alloc((size_t)MTOT * HIDDEN * sizeof(float));
    float* gi      = (float*)alloc((size_t)MTOT * 3 * HIDDEN * sizeof(float));
    float* gh      = (float*)alloc((size_t)MTOT * 3 * HIDDEN * sizeof(float));

    // per-call re-init (deterministic; no reliance on leftover ws state)
    hipMemsetAsync(cnt, 0, NUM_NODES * sizeof(int), stream);
    hipMemsetAsync(cursor, 0, NUM_NODES * sizeof(int), stream);
    hipMemsetAsync(state, 0, (size_t)MTOT * HIDDEN * sizeof(float), stream);

    // ---- one-time graph structure (CSR by dst, deterministic order) ----
    k_deg_int<<<(NUM_EDGES + 255) / 256, 256, 0, stream>>>(dst, cnt);
    k_node_norm<<<(NUM_NODES + 255) / 256, 256, 0, stream>>>(cnt, dinv, selfn);
    k_edge_norm<<<(NUM_EDGES + 255) / 256, 256, 0, stream>>>(src, dst, dinv, nrme);
    k_scan<<<1, 256, 0, stream>>>(cnt, rowptr);
    k_fill_csr<<<(NUM_EDGES + 255) / 256, 256, 0, stream>>>(src, dst, nrme, rowptr,
                                                            cursor, csr_eid, csr_src, csr_w);
    k_sort_csr<<<(NUM_NODES + 255) / 256, 256, 0, stream>>>(rowptr, csr_eid, csr_src, csr_w);
    // ---- layer-1 aggregation commuted past W1: xa = A_hat x, all 35 channels, once ----
    k_agg_x<<<MTOT / 8, 256, 0, stream>>>(x, rowptr, csr_src, csr_w, selfn, xa);

    const int eg   = (MTOT * HIDDEN) / 256;               // 40000 blocks
    const int g_h  = (MTOT / 16) * (HIDDEN / 16) / 4;     // GEMM -> [M,128]
    const int g_3h = (MTOT / 16) * (3 * HIDDEN / 16) / 4; // GEMM -> [M,384]

    for (int t = 0; t < LOOKBACK; ++t) {
        // layer 1: agg1+b1 = xa_t @ W1 + b1  (relu fused into next GEMM's A read)
        k_gcn1<<<eg, 256, 0, stream>>>(xa, W1, b1, bufA, t);
        // layer 2 matmul: h2 = relu(agg1) @ W2
        k_wmma_gemm<true><<<g_h, 128, 0, stream>>>(bufA, W2, nullptr, bufH,
            MTOT, HIDDEN, HIDDEN, HIDDEN, HIDDEN, 1, HIDDEN);
        // layer 2 aggregation (CSR gather, no atomics): agg2+b2
        k_aggregate<<<MTOT / 8, 256, 0, stream>>>(bufH, rowptr, csr_src, csr_w,
                                                  selfn, b2, bufA);
        // GRU: gi = relu(agg2) @ W_ih^T + b_ih ; gh = h @ W_hh^T + b_hh
        k_wmma_gemm<true><<<g_3h, 128, 0, stream>>>(bufA, W_ih, b_ih, gi,
            MTOT, 3 * HIDDEN, HIDDEN, HIDDEN, 1, HIDDEN, 3 * HIDDEN);
        k_wmma_gemm<false><<<g_3h, 128, 0, stream>>>(state, W_hh, b_hh, gh,
            MTOT, 3 * HIDDEN, HIDDEN, HIDDEN, 1, HIDDEN, 3 * HIDDEN);
        k_gru_gates<<<eg, 256, 0, stream>>>(gi, gh, state);
    }

    k_head<<<(MTOT + 255) / 256, 256, 0, stream>>>(state, W_out, b_out, out);
}